// XConv_2173253452339
// MI455X (gfx1250) — compile-verified
//
#include <hip/hip_runtime.h>

// ---------------------------------------------------------------------------
// PointCNN XConv pipeline for MI455X (gfx1250, wave32, WMMA).
// All matmuls use V_WMMA_F32_16X16X4_F32 (full fp32 precision; workload is
// bandwidth-bound at ~6.4 GFLOP vs ~350MB traffic, so fp32 WMMA is free).
// Six training-mode BatchNorms -> 6 global-reduction pass breaks with
// shfl-reduced float-atomic stats and tiny finalize kernels.
// ---------------------------------------------------------------------------

typedef float v2f __attribute__((ext_vector_type(2)));
typedef float v8f __attribute__((ext_vector_type(8)));

#define BB 8
#define NN 8192
#define MM 2048
#define KW 16
#define CC 128
#define HH 64
#define OO 256
#define CHN 192      // C + H
#define EPSI 1e-5f

__device__ __forceinline__ float eluf(float x) { return x > 0.f ? x : (__expf(x) - 1.f); }

// Sum across the 16 lanes of each wave half (C/D tile row lives on one half).
__device__ __forceinline__ float half16_sum(float v) {
  v += __shfl_xor(v, 1, 32);
  v += __shfl_xor(v, 2, 32);
  v += __shfl_xor(v, 4, 32);
  v += __shfl_xor(v, 8, 32);
  return v;
}

// D = A(16x4) * B(4x16) + C, fp32.  Fragment layout (per ISA 16x4 f32 table):
// A: lane%16 = row M, component c of v2f at half h holds K = k0 + 2*h + c.
// B: lane%16 = col N, component c at half h holds K = k0 + 2*h + c.
// C/D: vgpr r, half h -> row M = r + 8*h, col N = lane%16.
__device__ __forceinline__ v8f wmma4(v2f a, v2f b, v8f c) {
  return __builtin_amdgcn_wmma_f32_16x16x4_f32(false, a, false, b, (short)0, c,
                                               false, false);
}

// ---------------------------------------------------------------------------
// Stage 1: neighbor gather, center, f1 (3->64) + ELU, accumulate f1 BN stats.
// one thread per (b,m,k) pixel; stats wave-reduced before LDS atomics.
// ---------------------------------------------------------------------------
__global__ void k_gather_f1(const float* __restrict__ qp, const float* __restrict__ sp,
                            const float* __restrict__ w, const float* __restrict__ bias,
                            const int* __restrict__ nbr,
                            float* __restrict__ aligned, float* __restrict__ a1,
                            float* __restrict__ stat) {
  __shared__ float lw[HH * 3], lb[HH], lsum[HH], lsq[HH];
  int t = threadIdx.x;
  for (int i = t; i < HH * 3; i += 256) lw[i] = w[i];
  if (t < HH) { lb[t] = bias[t]; lsum[t] = 0.f; lsq[t] = 0.f; }
  __syncthreads();
  size_t pix = (size_t)blockIdx.x * 256 + t;     // 0 .. B*M*K-1
  int b = (int)(pix / (MM * KW));
  int rem = (int)(pix - (size_t)b * MM * KW);
  int m = rem / KW;
  int idx = nbr[pix];
  float p0 = sp[(b * 3 + 0) * NN + idx] - qp[(b * 3 + 0) * MM + m];
  float p1 = sp[(b * 3 + 1) * NN + idx] - qp[(b * 3 + 1) * MM + m];
  float p2 = sp[(b * 3 + 2) * NN + idx] - qp[(b * 3 + 2) * MM + m];
  aligned[pix * 3 + 0] = p0;
  aligned[pix * 3 + 1] = p1;
  aligned[pix * 3 + 2] = p2;
  float* out = a1 + pix * HH;
  int lane = t & 31;
  for (int o = 0; o < HH; o++) {
    float e = eluf(lw[o * 3] * p0 + lw[o * 3 + 1] * p1 + lw[o * 3 + 2] * p2 + lb[o]);
    out[o] = e;
    float s1 = half16_sum(e);
    float s2 = half16_sum(e * e);
    s1 += __shfl_xor(s1, 16, 32);
    s2 += __shfl_xor(s2, 16, 32);
    if (lane == 0) { atomicAdd(&lsum[o], s1); atomicAdd(&lsq[o], s2); }
  }
  __syncthreads();
  if (t < HH) { atomicAdd(&stat[t], lsum[t]); atomicAdd(&stat[HH + t], lsq[t]); }
}

// ---------------------------------------------------------------------------
// BN finalize: scale = g * rsqrt(var + eps), shift = be - mean * scale.
// ---------------------------------------------------------------------------
__global__ void k_finalize(const float* __restrict__ ssum, const float* __restrict__ ssq,
                           const float* __restrict__ g, const float* __restrict__ be,
                           float* __restrict__ sc, float* __restrict__ sh,
                           float inv_count, int nch) {
  int c = blockIdx.x * blockDim.x + threadIdx.x;
  if (c >= nch) return;
  float mean = ssum[c] * inv_count;
  float var = ssq[c] * inv_count - mean * mean;
  float rs = rsqrtf(var + EPSI);
  float s = g[c] * rs;
  sc[c] = s;
  sh[c] = be[c] - mean * s;
}

// ---------------------------------------------------------------------------
// Stage 2: f2 (64->64) with WMMA; one wave per 16-pixel group (= one (b,m)).
// Applies f1 BN on the fly; ELU; in-place a1 -> a2; accumulates f2 stats.
// ---------------------------------------------------------------------------
__global__ void k_f2(const float* __restrict__ w, const float* __restrict__ bias,
                     const float* __restrict__ sc1, const float* __restrict__ sh1,
                     float* __restrict__ a, float* __restrict__ stat) {
  __shared__ float lw[HH * HH];
  __shared__ float lb[HH], lsc[HH], lsh[HH], lsum[HH], lsq[HH];
  int t = threadIdx.x;
  for (int i = t; i < HH * HH; i += 256) lw[i] = w[i];
  if (t < HH) { lb[t] = bias[t]; lsc[t] = sc1[t]; lsh[t] = sh1[t]; lsum[t] = 0.f; lsq[t] = 0.f; }
  __syncthreads();
  int wave = t >> 5, lane = t & 31, half = lane >> 4, l16 = lane & 15;
  size_t grp = (size_t)blockIdx.x * 8 + wave;
  float* arow = a + (grp * 16 + l16) * HH;
  v8f acc[4] = {};
  for (int s = 0; s < 16; s++) {
    int ch = 4 * s + 2 * half;
    v2f bf;
    bf.x = arow[ch] * lsc[ch] + lsh[ch];
    bf.y = arow[ch + 1] * lsc[ch + 1] + lsh[ch + 1];
    for (int ti = 0; ti < 4; ti++) {
      const float* wr = &lw[(ti * 16 + l16) * HH + ch];
      v2f af; af.x = wr[0]; af.y = wr[1];
      acc[ti] = wmma4(af, bf, acc[ti]);
    }
  }
  for (int ti = 0; ti < 4; ti++) {
    for (int r = 0; r < 8; r++) {
      int o = ti * 16 + r + 8 * half;
      float e = eluf(acc[ti][r] + lb[o]);
      arow[o] = e;                                  // in-place: all reads done
      float s1 = half16_sum(e);
      float s2 = half16_sum(e * e);
      if (l16 == 0) { atomicAdd(&lsum[o], s1); atomicAdd(&lsq[o], s2); }
    }
  }
  __syncthreads();
  if (t < HH) { atomicAdd(&stat[t], lsum[t]); atomicAdd(&stat[HH + t], lsq[t]); }
}

// ---------------------------------------------------------------------------
// Stage 3: stem [256 x 48] @ [48 x 16m] with WMMA; weights pre-permuted to
// contraction index j = k*3+c in LDS.  ELU; stats.
// ---------------------------------------------------------------------------
__global__ void k_stem(const float* __restrict__ w, const float* __restrict__ bias,
                       const float* __restrict__ aligned,
                       float* __restrict__ t0, float* __restrict__ stat) {
  __shared__ float lw[256 * 48];   // 48KB
  __shared__ float lb[256], lsum[256], lsq[256];
  int t = threadIdx.x;
  for (int i = t; i < 256 * 48; i += 256) {
    int o = i / 48, j = i % 48;
    lw[i] = w[o * 48 + (j % 3) * 16 + (j / 3)];   // stem_w[o][c][k], j = k*3+c
  }
  lb[t] = bias[t]; lsum[t] = 0.f; lsq[t] = 0.f;
  __syncthreads();
  int wave = t >> 5, lane = t & 31, half = lane >> 4, l16 = lane & 15;
  size_t bm = ((size_t)blockIdx.x * 8 + wave) * 16 + l16;
  const float* arow = aligned + bm * 48;
  v2f bfr[12];
  for (int s = 0; s < 12; s++) {
    int j = 4 * s + 2 * half;
    bfr[s].x = arow[j]; bfr[s].y = arow[j + 1];
  }
  float* orow = t0 + bm * 256;
  for (int ti = 0; ti < 16; ti++) {
    v8f acc = {};
    for (int s = 0; s < 12; s++) {
      int j = 4 * s + 2 * half;
      v2f af;
      af.x = lw[(ti * 16 + l16) * 48 + j];
      af.y = lw[(ti * 16 + l16) * 48 + j + 1];
      acc = wmma4(af, bfr[s], acc);
    }
    for (int r = 0; r < 8; r++) {
      int o = ti * 16 + r + 8 * half;
      float e = eluf(acc[r] + lb[o]);
      orow[o] = e;
      float s1 = half16_sum(e);
      float s2 = half16_sum(e * e);
      if (l16 == 0) { atomicAdd(&lsum[o], s1); atomicAdd(&lsq[o], s2); }
    }
  }
  __syncthreads();
  atomicAdd(&stat[t], lsum[t]);
  atomicAdd(&stat[256 + t], lsq[t]);
}

// ---------------------------------------------------------------------------
// Stage 4: grouped conv (16 groups of 16x16) with WMMA.  Applies previous
// stage's BN (optionally channel-swapped due to the XReshape transpose),
// stores with the transpose baked in (out channel = p*16+g), stats on g*16+p.
// ---------------------------------------------------------------------------
__global__ void k_xconv(const float* __restrict__ w, const float* __restrict__ bias,
                        const float* __restrict__ scIn, const float* __restrict__ shIn,
                        int swapIn, const float* __restrict__ tin,
                        float* __restrict__ tout, float* __restrict__ stat) {
  __shared__ float lw[16 * 16 * 16];
  __shared__ float lb[256], lsc[256], lsh[256], lsum[256], lsq[256];
  int t = threadIdx.x;
  for (int i = t; i < 4096; i += 256) lw[i] = w[i];
  {
    int src = swapIn ? ((t & 15) * 16 + (t >> 4)) : t;
    lb[t] = bias[t]; lsc[t] = scIn[src]; lsh[t] = shIn[src];
    lsum[t] = 0.f; lsq[t] = 0.f;
  }
  __syncthreads();
  int wave = t >> 5, lane = t & 31, half = lane >> 4, l16 = lane & 15;
  size_t bm = ((size_t)blockIdx.x * 8 + wave) * 16 + l16;
  const float* irow = tin + bm * 256;
  float* orow = tout + bm * 256;
  for (int g = 0; g < 16; g++) {
    v8f acc = {};
    for (int s = 0; s < 4; s++) {
      int j = 4 * s + 2 * half;
      int ch = g * 16 + j;
      v2f bf;
      bf.x = irow[ch] * lsc[ch] + lsh[ch];
      bf.y = irow[ch + 1] * lsc[ch + 1] + lsh[ch + 1];
      v2f af;
      af.x = lw[g * 256 + l16 * 16 + j];       // w[g][p=l16][j]
      af.y = lw[g * 256 + l16 * 16 + j + 1];
      acc = wmma4(af, bf, acc);
    }
    for (int r = 0; r < 8; r++) {
      int p = r + 8 * half;
      int obn = g * 16 + p;                    // BN channel
      float e = eluf(acc[r] + lb[obn]);
      orow[p * 16 + g] = e;                    // transposed storage
      float s1 = half16_sum(e);
      float s2 = half16_sum(e * e);
      if (l16 == 0) { atomicAdd(&lsum[obn], s1); atomicAdd(&lsq[obn], s2); }
    }
  }
  __syncthreads();
  atomicAdd(&stat[t], lsum[t]);
  atomicAdd(&stat[256 + t], lsq[t]);
}

// ---------------------------------------------------------------------------
// Stage 5: per-(b,m) qf = nf[192x16] @ X[16x16] via WMMA, then depthwise
// reduction dw[c] = sum_l qf[c][l] * dw_w[c][l] via shfl_xor.  nf channels
// 0..127 gathered from s_feats, 128..191 = f2 output with BN on the fly.
// X gets x2's BN with channel swap.  One wave per (b,m).
// ---------------------------------------------------------------------------
__global__ void k_dw(const float* __restrict__ sfeat, const int* __restrict__ nbr,
                     const float* __restrict__ a2, const float* __restrict__ scF2,
                     const float* __restrict__ shF2, const float* __restrict__ t2,
                     const float* __restrict__ scX2, const float* __restrict__ shX2,
                     const float* __restrict__ dww, float* __restrict__ dwout) {
  __shared__ float ldw[CHN * 16];
  __shared__ float lscF[HH], lshF[HH], lscX[256], lshX[256];
  int t = threadIdx.x;
  for (int i = t; i < CHN * 16; i += 256) ldw[i] = dww[i];
  if (t < HH) { lscF[t] = scF2[t]; lshF[t] = shF2[t]; }
  { int src = (t & 15) * 16 + (t >> 4); lscX[t] = scX2[src]; lshX[t] = shX2[src]; }
  __syncthreads();
  int wave = t >> 5, lane = t & 31, half = lane >> 4, l16 = lane & 15;
  size_t bm = (size_t)blockIdx.x * 8 + wave;
  int b = (int)(bm / MM);
  const float* trow = t2 + bm * 256;
  v2f bf[4];                                    // X[k][l], rows k, cols l=l16
  for (int s = 0; s < 4; s++) {
    int k = 4 * s + 2 * half;
    int ch0 = l16 * 16 + k;                     // stored channel = l*16+k
    bf[s].x = trow[ch0] * lscX[ch0] + lshX[ch0];
    bf[s].y = trow[ch0 + 1] * lscX[ch0 + 1] + lshX[ch0 + 1];
  }
  const int* nrow = nbr + bm * KW;
  const float* a2row = a2 + bm * KW * HH;
  const float* sfb = sfeat + (size_t)b * CC * NN;
  float* drow = dwout + bm * CHN;
  for (int ti = 0; ti < 12; ti++) {
    int c = ti * 16 + l16;                      // uniform c<CC per tile (128=8*16)
    v8f acc = {};
    for (int s = 0; s < 4; s++) {
      int k = 4 * s + 2 * half;
      v2f af;
      if (c < CC) {
        af.x = sfb[(size_t)c * NN + nrow[k]];
        af.y = sfb[(size_t)c * NN + nrow[k + 1]];
      } else {
        int hc = c - CC;
        af.x = a2row[k * HH + hc] * lscF[hc] + lshF[hc];
        af.y = a2row[(k + 1) * HH + hc] * lscF[hc] + lshF[hc];
      }
      acc = wmma4(af, bf[s], acc);
    }
    for (int r = 0; r < 8; r++) {
      int cc = ti * 16 + r + 8 * half;
      float v = half16_sum(acc[r] * ldw[cc * 16 + l16]);
      if (l16 == 0) drow[cc] = v;
    }
  }
}

// ---------------------------------------------------------------------------
// Stage 6: pointwise [256 x 192] @ [192 x 16m] with WMMA; ELU; sep stats;
// writes pre-BN values to d_out (normalized in place afterwards).
// pw_w (192KB) stays in L2; next tile's row is prefetched.
// ---------------------------------------------------------------------------
__global__ void k_pw(const float* __restrict__ pww, const float* __restrict__ pwb,
                     const float* __restrict__ dwin, float* __restrict__ dout,
                     float* __restrict__ stat) {
  __shared__ float lb[256], lsum[256], lsq[256];
  int t = threadIdx.x;
  lb[t] = pwb[t]; lsum[t] = 0.f; lsq[t] = 0.f;
  __syncthreads();
  int wave = t >> 5, lane = t & 31, half = lane >> 4, l16 = lane & 15;
  size_t mg = (size_t)blockIdx.x * 8 + wave;
  size_t bm = mg * 16 + l16;
  const float* drow = dwin + bm * CHN;
  v2f bfr[48];
  for (int s = 0; s < 48; s++) {
    int c = 4 * s + 2 * half;
    bfr[s].x = drow[c]; bfr[s].y = drow[c + 1];
  }
  size_t b = (mg * 16) / MM;                    // uniform in wave (M%16==0)
  size_t m0 = (mg * 16) % MM;
  float* ob = dout + b * (size_t)OO * MM;
  __builtin_prefetch(pww + (size_t)l16 * CHN, 0, 1);
  for (int ti = 0; ti < 16; ti++) {
    v8f acc = {};
    const float* wr = pww + (size_t)(ti * 16 + l16) * CHN;
    if (ti < 15) __builtin_prefetch(pww + (size_t)((ti + 1) * 16 + l16) * CHN, 0, 1);
    for (int s = 0; s < 48; s++) {
      int c = 4 * s + 2 * half;
      v2f af; af.x = wr[c]; af.y = wr[c + 1];
      acc = wmma4(af, bfr[s], acc);
    }
    for (int r = 0; r < 8; r++) {
      int o = ti * 16 + r + 8 * half;
      float e = eluf(acc[r] + lb[o]);
      ob[(size_t)o * MM + m0 + l16] = e;
      float s1 = half16_sum(e);
      float s2 = half16_sum(e * e);
      if (l16 == 0) { atomicAdd(&lsum[o], s1); atomicAdd(&lsq[o], s2); }
    }
  }
  __syncthreads();
  atomicAdd(&stat[t], lsum[t]);
  atomicAdd(&stat[256 + t], lsq[t]);
}

__global__ void k_outnorm(float* __restrict__ out, const float* __restrict__ sc,
                          const float* __restrict__ sh) {
  size_t i = (size_t)blockIdx.x * 256 + threadIdx.x;
  int ch = (int)((i / MM) % OO);
  out[i] = out[i] * sc[ch] + sh[ch];
}

// ---------------------------------------------------------------------------
extern "C" void kernel_launch(void* const* d_in, const int* in_sizes, int n_in,
                              void* d_out, int out_size, void* d_ws, size_t ws_size,
                              hipStream_t stream) {
  (void)in_sizes; (void)n_in; (void)out_size; (void)ws_size;
  const float* qp   = (const float*)d_in[0];
  const float* sp   = (const float*)d_in[1];
  const float* sf   = (const float*)d_in[2];
  const float* f1w  = (const float*)d_in[3];
  const float* f1b  = (const float*)d_in[4];
  const float* f1g  = (const float*)d_in[5];
  const float* f1be = (const float*)d_in[6];
  const float* f2w  = (const float*)d_in[7];
  const float* f2b  = (const float*)d_in[8];
  const float* f2g  = (const float*)d_in[9];
  const float* f2be = (const float*)d_in[10];
  const float* stw  = (const float*)d_in[11];
  const float* stb  = (const float*)d_in[12];
  const float* stg  = (const float*)d_in[13];
  const float* stbe = (const float*)d_in[14];
  const float* x1w  = (const float*)d_in[15];
  const float* x1b  = (const float*)d_in[16];
  const float* x1g  = (const float*)d_in[17];
  const float* x1be = (const float*)d_in[18];
  const float* x2w  = (const float*)d_in[19];
  const float* x2b  = (const float*)d_in[20];
  const float* x2g  = (const float*)d_in[21];
  const float* x2be = (const float*)d_in[22];
  const float* dww  = (const float*)d_in[23];
  const float* pww  = (const float*)d_in[24];
  const float* pwb  = (const float*)d_in[25];
  const float* sepg = (const float*)d_in[26];
  const float* sepbe= (const float*)d_in[27];
  const int*   nbr  = (const int*)d_in[28];
  float* out = (float*)d_out;
  float* ws  = (float*)d_ws;

  // workspace layout (floats)
  float* st    = ws;              // 2304: stats sum/sq per stage
  float* f1sc = ws + 2304, *f1sh = ws + 2368;
  float* f2sc = ws + 2432, *f2sh = ws + 2496;
  float* stsc = ws + 2560, *stsh = ws + 2816;
  float* x1sc = ws + 3072, *x1sh = ws + 3328;
  float* x2sc = ws + 3584, *x2sh = ws + 3840;
  float* sesc = ws + 4096, *sesh = ws + 4352;
  float* aligned = ws + 4608;                       //  786432
  float* a   = aligned + (size_t)786432;            // 16777216 (a1 then a2 in place)
  float* t0  = a  + (size_t)16777216;               //  4194304
  float* t1  = t0 + (size_t)4194304;
  float* t2  = t1 + (size_t)4194304;
  float* dwb = t2 + (size_t)4194304;                //  3145728

  hipMemsetAsync(ws, 0, 2304 * sizeof(float), stream);

  const float invBMK = 1.f / (float)(BB * MM * KW);  // 1/262144
  const float invBM  = 1.f / (float)(BB * MM);       // 1/16384

  k_gather_f1<<<1024, 256, 0, stream>>>(qp, sp, f1w, f1b, nbr, aligned, a, st + 0);
  k_finalize<<<1, 64, 0, stream>>>(st + 0, st + 64, f1g, f1be, f1sc, f1sh, invBMK, 64);
  k_f2<<<2048, 256, 0, stream>>>(f2w, f2b, f1sc, f1sh, a, st + 128);
  k_finalize<<<1, 64, 0, stream>>>(st + 128, st + 192, f2g, f2be, f2sc, f2sh, invBMK, 64);
  k_stem<<<128, 256, 0, stream>>>(stw, stb, aligned, t0, st + 256);
  k_finalize<<<1, 256, 0, stream>>>(st + 256, st + 512, stg, stbe, stsc, stsh, invBM, 256);
  k_xconv<<<128, 256, 0, stream>>>(x1w, x1b, stsc, stsh, 0, t0, t1, st + 768);
  k_finalize<<<1, 256, 0, stream>>>(st + 768, st + 1024, x1g, x1be, x1sc, x1sh, invBM, 256);
  k_xconv<<<128, 256, 0, stream>>>(x2w, x2b, x1sc, x1sh, 1, t1, t2, st + 1280);
  k_finalize<<<1, 256, 0, stream>>>(st + 1280, st + 1536, x2g, x2be, x2sc, x2sh, invBM, 256);
  k_dw<<<2048, 256, 0, stream>>>(sf, nbr, a, f2sc, f2sh, t2, x2sc, x2sh, dww, dwb);
  k_pw<<<128, 256, 0, stream>>>(pww, pwb, dwb, out, st + 1792);
  k_finalize<<<1, 256, 0, stream>>>(st + 1792, st + 2048, sepg, sepbe, sesc, sesh, invBM, 256);
  k_outnorm<<<(BB * OO * MM) / 256, 256, 0, stream>>>(out, sesc, sesh);
}